// CAPE_43525198578167
// MI455X (gfx1250) — compile-verified
//
#include <hip/hip_runtime.h>

typedef __attribute__((ext_vector_type(2))) float v2f;
typedef __attribute__((ext_vector_type(8))) float v8f;

#define NSAMP 10
#define DIM   64

// One wave32 handles 16 consecutive batch elements.
// target_loss and the 10 negative losses are computed as diagonals of
// 16x16 f32 WMMA tiles (V_WMMA_F32_16X16X4_F32, K accumulated 4 at a time).
__global__ __launch_bounds__(256) void sgns_wmma_kernel(
    const float* __restrict__ emb_in,
    const float* __restrict__ emb_out,
    const int*   __restrict__ target,
    const int*   __restrict__ context,
    const int*   __restrict__ negs,
    float*       __restrict__ out_target,   // [B]
    float*       __restrict__ out_neg)      // [B, NSAMP]
{
    const int lane = threadIdx.x & 31;
    const int wave = threadIdx.x >> 5;
    const int b0   = (blockIdx.x * (blockDim.x >> 5) + wave) << 4;

    const int m    = lane & 15;          // batch slot this lane loads for
    const int bm   = b0 + m;
    const int koff = (lane >> 4) << 1;   // 0 for lanes 0-15, 2 for lanes 16-31

    // ---- gather indices (issued early, latencies overlap) ----
    const int t_idx = target[bm];
    const int c_idx = context[bm];
    int nidx[NSAMP];
#pragma unroll
    for (int j = 0; j < NSAMP; ++j)
        nidx[j] = negs[(size_t)bm * NSAMP + j];

    // ---- preload A = e_in rows, in WMMA 16x4 chunk layout ----
    // A[kk] holds A-matrix chunk K = 4*kk .. 4*kk+3:
    //   lane L<16 : e_in[b0+L][4kk+0..1], lane L>=16 : e_in[b0+L-16][4kk+2..3]
    const float* ain = emb_in + (size_t)t_idx * DIM + koff;
    v2f A[16];
#pragma unroll
    for (int kk = 0; kk < 16; ++kk)
        A[kk] = *(const v2f*)(ain + 4 * kk);

    // Diagonal D[n][n] lives in: lanes 0-7 -> acc[lane]; lanes 24-31 -> acc[lane-24]
    const bool writer = (lane < 8) || (lane >= 24);
    const int  n      = lane & 15;   // output column for writer lanes
    const int  r      = lane & 7;    // accumulator VGPR row holding D[n][n]

    // ---- target loss: diag( e_in_block x e_out_block^T ) ----
    {
        const float* bptr = emb_out + (size_t)c_idx * DIM + koff;
        v8f acc = {0.f, 0.f, 0.f, 0.f, 0.f, 0.f, 0.f, 0.f};
#pragma unroll
        for (int kk = 0; kk < 16; ++kk) {
            v2f Bv = *(const v2f*)(bptr + 4 * kk);
            acc = __builtin_amdgcn_wmma_f32_16x16x4_f32(
                false, A[kk], false, Bv, (short)0, acc, false, false);
        }
        if (writer)
            out_target[b0 + n] = acc[r];
    }

    // ---- negative losses: diag( e_in_block x neg_j_block^T ), negated ----
#pragma unroll 2
    for (int j = 0; j < NSAMP; ++j) {
        const float* bptr = emb_out + (size_t)nidx[j] * DIM + koff;
        v8f acc = {0.f, 0.f, 0.f, 0.f, 0.f, 0.f, 0.f, 0.f};
#pragma unroll
        for (int kk = 0; kk < 16; ++kk) {
            v2f Bv = *(const v2f*)(bptr + 4 * kk);
            acc = __builtin_amdgcn_wmma_f32_16x16x4_f32(
                false, A[kk], false, Bv, (short)0, acc, false, false);
        }
        if (writer)
            out_neg[(size_t)(b0 + n) * NSAMP + j] = -acc[r];
    }
}

extern "C" void kernel_launch(void* const* d_in, const int* in_sizes, int n_in,
                              void* d_out, int out_size, void* d_ws, size_t ws_size,
                              hipStream_t stream) {
    const float* emb_in  = (const float*)d_in[0];   // [POI, 64] f32
    const float* emb_out = (const float*)d_in[1];   // [POI, 64] f32
    const int*   target  = (const int*)d_in[2];     // [B]
    const int*   context = (const int*)d_in[3];     // [B]
    const int*   negs    = (const int*)d_in[4];     // [B, 10]
    const int    B       = in_sizes[2];

    float* out_target = (float*)d_out;          // [B]
    float* out_neg    = (float*)d_out + B;      // [B, 10] (flattened [B,10,1])

    const int waves_per_block = 8;              // 256 threads
    const int b_per_block     = 16 * waves_per_block;  // 128
    dim3 block(32 * waves_per_block);
    dim3 grid(B / b_per_block);

    hipLaunchKernelGGL(sgns_wmma_kernel, grid, block, 0, stream,
                       emb_in, emb_out, target, context, negs,
                       out_target, out_neg);
}